// MultiBoxLoss_41626823033145
// MI455X (gfx1250) — compile-verified
//
#include <hip/hip_runtime.h>
#include <hip/hip_bf16.h>
#include <math.h>

#define Bc   16
#define Pc   16384
#define Oc   16
#define Cc   21
#define Hc   256
#define Wc   256
#define HWc  (Hc*Wc)

typedef float v2f __attribute__((ext_vector_type(2)));
typedef float v8f __attribute__((ext_vector_type(8)));

// ---------------- workspace layout (bytes) ----------------
#define OFF_BEST   0u                      // 256 * u64          = 2048
#define OFF_NPOS   2048u                   // 16  * int          = 64
#define OFF_PHARD  2112u                   // 16  * float        = 64
#define OFF_PLOC   2176u                   // 1024 * float       = 4096
#define OFF_PCE    6272u                   // 1024 * float       = 4096
#define OFF_PMASK  10368u                  // 1024 * float       = 4096
#define OFF_OBJ    14464u                  // B*P ints           = 1 MB
#define OFF_OVL    (OFF_OBJ  + Bc*Pc*4u)   // B*P floats         = 1 MB
#define OFF_CENEG  (OFF_OVL  + Bc*Pc*4u)   // B*P floats         = 1 MB

// ---------------- helpers ----------------
__device__ __forceinline__ float blockReduceSum(float v, float* sbuf) {
  const int lane = threadIdx.x & 31;
  const int wid  = threadIdx.x >> 5;
#pragma unroll
  for (int o = 16; o; o >>= 1) v += __shfl_down(v, o, 32);
  if (lane == 0) sbuf[wid] = v;
  __syncthreads();
  const int nw = blockDim.x >> 5;
  v = ((int)threadIdx.x < nw) ? sbuf[threadIdx.x] : 0.f;
  if (wid == 0) {
#pragma unroll
    for (int o = 16; o; o >>= 1) v += __shfl_down(v, o, 32);
  }
  return v;  // valid on thread 0
}

// ---------------- kernel 1: init ----------------
__global__ void k_init(unsigned long long* best, int* nPos) {
  int t = threadIdx.x;                 // 256 threads
  best[t] = 0ull;
  if (t < Bc) nPos[t] = 0;
}

// ---------------- kernel 2: IoU + assignment ----------------
__global__ __launch_bounds__(256) void k_iou(
    const float4* __restrict__ boxes4,     // (B,O) xyxy
    const float4* __restrict__ priors4,    // (P)   cxcy
    unsigned long long* __restrict__ best, // (B*O) packed (iou_bits<<32)|~prior
    int*   __restrict__ objArr,            // (B*P)
    float* __restrict__ ovlArr) {          // (B*P)
  __shared__ float4 sbox[Oc];
  __shared__ unsigned long long swk[8];
  const int b = blockIdx.y;
  const int p = blockIdx.x * 256 + threadIdx.x;
  if (threadIdx.x < Oc) sbox[threadIdx.x] = boxes4[b * Oc + threadIdx.x];
  __syncthreads();
  const float4 pr = priors4[p];
  const float px1 = pr.x - pr.z * 0.5f, py1 = pr.y - pr.w * 0.5f;
  const float px2 = pr.x + pr.z * 0.5f, py2 = pr.y + pr.w * 0.5f;
  const float pa = (px2 - px1) * (py2 - py1);

  float bestOv = -1.f; int bestO = 0;
  const int lane = threadIdx.x & 31, wid = threadIdx.x >> 5;
  for (int o = 0; o < Oc; ++o) {
    float4 bx = sbox[o];
    float lox = fmaxf(bx.x, px1), loy = fmaxf(bx.y, py1);
    float hix = fminf(bx.z, px2), hiy = fminf(bx.w, py2);
    float iw = fmaxf(hix - lox, 0.f), ih = fmaxf(hiy - loy, 0.f);
    float inter = iw * ih;
    float a1 = (bx.z - bx.x) * (bx.w - bx.y);
    float iou = inter / (a1 + pa - inter);
    if (iou > bestOv) { bestOv = iou; bestO = o; }
    // per-object block-wide argmax over priors (packed key, tie -> lower prior)
    unsigned long long key =
        ((unsigned long long)__float_as_uint(iou) << 32) |
        (unsigned long long)(0xFFFFFFFFu - (unsigned)p);
#pragma unroll
    for (int off = 16; off; off >>= 1) {
      unsigned long long other = __shfl_down(key, off, 32);
      if (other > key) key = other;
    }
    if (lane == 0) swk[wid] = key;
    __syncthreads();
    if (threadIdx.x == 0) {
      unsigned long long k2 = swk[0];
#pragma unroll
      for (int i = 1; i < 8; ++i) if (swk[i] > k2) k2 = swk[i];
      atomicMax(&best[b * Oc + o], k2);
    }
    __syncthreads();
  }
  objArr[b * Pc + p] = bestO;
  ovlArr[b * Pc + p] = bestOv;
}

// ---------------- kernel 3: force winners ----------------
__global__ void k_winner(const unsigned long long* __restrict__ best,
                         int* __restrict__ objArr, float* __restrict__ ovlArr) {
  int t = threadIdx.x;                 // 256 threads = B*O
  if (t < Bc * Oc) {
    int b = t / Oc, o = t % Oc;
    unsigned q = 0xFFFFFFFFu - (unsigned)(best[t] & 0xFFFFFFFFull);
    objArr[b * Pc + q] = o;
    ovlArr[b * Pc + q] = 1.0f;
  }
}

// ---------------- kernel 4: per-prior loc + CE ----------------
__global__ __launch_bounds__(256) void k_prior_loss(
    const float4* __restrict__ plocs4,   // (B,P)
    const float*  __restrict__ pscores,  // (B,P,C)
    const float4* __restrict__ boxes4,   // (B,O)
    const float4* __restrict__ priors4,  // (P)
    const int*    __restrict__ labels,   // (B,O)
    const int*    __restrict__ objArr,
    const float*  __restrict__ ovlArr,
    float* __restrict__ ceNeg,           // (B,P)
    int*   __restrict__ nPos,            // (B)
    float* __restrict__ pLoc,            // (1024)
    float* __restrict__ pCe) {           // (1024)
  __shared__ float sred[8];
  const int b = blockIdx.y;
  const int p = blockIdx.x * 256 + threadIdx.x;
  const int obj = objArr[b * Pc + p];
  const float ov = ovlArr[b * Pc + p];
  int lab = labels[b * Oc + obj];
  if (ov < 0.5f) lab = 0;
  const bool pos = (lab != 0);

  // encoded true locs
  const float4 bx = boxes4[b * Oc + obj];
  float cx = (bx.x + bx.z) * 0.5f, cy = (bx.y + bx.w) * 0.5f;
  float w = bx.z - bx.x, h = bx.w - bx.y;
  const float4 pr = priors4[p];
  float gx = (cx - pr.x) / (pr.z * 0.1f);
  float gy = (cy - pr.y) / (pr.w * 0.1f);
  float gw = __logf(w / pr.z) * 5.f;
  float gh = __logf(h / pr.w) * 5.f;
  const float4 pl = plocs4[(size_t)b * Pc + p];
  float dsum = fabsf(pl.x - gx) + fabsf(pl.y - gy) +
               fabsf(pl.z - gw) + fabsf(pl.w - gh);
  dsum = pos ? dsum : 0.f;

  // CE over 21 classes, single pass, registers
  const float* sc = pscores + ((size_t)(b * Pc + p)) * Cc;
  float v[Cc];
  float m = -3.4e38f, st = 0.f;
#pragma unroll
  for (int c = 0; c < Cc; ++c) {
    float x = sc[c];
    v[c] = x;
    m = fmaxf(m, x);
    if (c == lab) st = x;
  }
  float se = 0.f;
#pragma unroll
  for (int c = 0; c < Cc; ++c) se += __expf(v[c] - m);
  float ce = m + __logf(se) - st;
  ceNeg[b * Pc + p] = pos ? 0.f : ce;

  float rl = blockReduceSum(dsum, sred);
  __syncthreads();
  float rc = blockReduceSum(pos ? ce : 0.f, sred);
  __syncthreads();
  float rn = blockReduceSum(pos ? 1.f : 0.f, sred);
  const int blk = blockIdx.y * gridDim.x + blockIdx.x;
  if (threadIdx.x == 0) {
    pLoc[blk] = rl;
    pCe[blk]  = rc;
    atomicAdd(&nPos[b], (int)rn);
  }
}

// ---------------- kernel 5: hard-negative mining (LDS bitonic sort) ----------------
__global__ __launch_bounds__(1024) void k_hardneg(
    const float* __restrict__ ceNeg, const int* __restrict__ nPos,
    float* __restrict__ pHard) {
  __shared__ float s[Pc];              // 64 KB of the 320 KB WGP LDS
  __shared__ float sred[32];
  const int b = blockIdx.x;
  // b128 global loads -> ds_store_b128
  const float4* src = (const float4*)(ceNeg + (size_t)b * Pc);
  float4* s4 = (float4*)s;
  for (int i = threadIdx.x; i < Pc / 4; i += 1024) s4[i] = src[i];
  __syncthreads();
  for (unsigned k = 2; k <= (unsigned)Pc; k <<= 1) {
    for (unsigned j = k >> 1; j > 0; j >>= 1) {
      for (unsigned i = threadIdx.x; i < (unsigned)Pc; i += 1024) {
        unsigned ix = i ^ j;
        if (ix > i) {
          float a = s[i], c2 = s[ix];
          bool descend = ((i & k) == 0);
          bool doswap = descend ? (a < c2) : (a > c2);
          if (doswap) { s[i] = c2; s[ix] = a; }
        }
      }
      __syncthreads();
    }
  }
  int K = 3 * nPos[b];
  if (K > Pc) K = Pc;
  float acc = 0.f;
  for (int i = threadIdx.x; i < K; i += 1024) acc += s[i];
  float r = blockReduceSum(acc, sred);
  if (threadIdx.x == 0) pHard[b] = r;
}

// ---------------- kernel 6: mask CE (streams 88 MB once, 2 px/thread b64) ----------------
__global__ __launch_bounds__(512) void k_mask(
    const float* __restrict__ pm,     // (B,C,H,W)
    const int*   __restrict__ masks,  // (B,1,H,W)
    float* __restrict__ pMask) {      // (1024)
  __shared__ int   stgt[1024];
  __shared__ float sred[16];
  const int b    = blockIdx.y;
  const int pix0 = (blockIdx.x * 512 + threadIdx.x) * 2;
  const size_t idx0 = (size_t)b * HWc + pix0;

  // gfx1250 async global->LDS stage of the target tile (ASYNCcnt-tracked)
  {
    unsigned long long gaddr = (unsigned long long)(const void*)(masks + idx0);
    unsigned ldsaddr = (unsigned)(unsigned long long)(void*)&stgt[2 * threadIdx.x];
    asm volatile("global_load_async_to_lds_b64 %0, %1, off"
                 :: "v"(ldsaddr), "v"(gaddr) : "memory");
    asm volatile("s_wait_asynccnt 0" ::: "memory");
  }
  __syncthreads();
  const int t0 = stgt[2 * threadIdx.x];
  const int t1 = stgt[2 * threadIdx.x + 1];
  const bool va = (t0 != 255), vb = (t1 != 255);
  const int ta = va ? t0 : 0, tb = vb ? t1 : 0;

  const float2* base =
      (const float2*)(pm + (size_t)b * Cc * HWc + pix0);
  float2 v[Cc];
  float ma = -3.4e38f, mb = -3.4e38f, xa = 0.f, xb = 0.f;
#pragma unroll
  for (int c = 0; c < Cc; ++c) {
    float2 x = base[(size_t)c * (HWc / 2)];  // coalesced b64 per lane
    v[c] = x;
    ma = fmaxf(ma, x.x);
    mb = fmaxf(mb, x.y);
    if (c == ta) xa = x.x;
    if (c == tb) xb = x.y;
  }
  float sa = 0.f, sb = 0.f;
#pragma unroll
  for (int c = 0; c < Cc; ++c) {
    sa += __expf(v[c].x - ma);
    sb += __expf(v[c].y - mb);
  }
  float ce = (va ? (ma + __logf(sa) - xa) : 0.f) +
             (vb ? (mb + __logf(sb) - xb) : 0.f);
  float r = blockReduceSum(ce, sred);
  if (threadIdx.x == 0) pMask[blockIdx.y * gridDim.x + blockIdx.x] = r;
}

// ---------------- kernel 7: final combine (WMMA fp32 reduction) ----------------
// One wave sums a 1024-float array: 16 iterations of V_WMMA_F32_16X16X4_F32,
// A = 16x4 data chunk, B = ones => D rows accumulate row sums through C.
__device__ __forceinline__ float wmmaSum1024(const float* part, int lane) {
  v8f c = {};
  v2f ones; ones.x = 1.f; ones.y = 1.f;
  const int base = (lane < 16) ? 4 * lane : 4 * (lane - 16) + 2;
  for (int t = 0; t < 16; ++t) {
    v2f a;
    a.x = part[t * 64 + base];
    a.y = part[t * 64 + base + 1];
    c = __builtin_amdgcn_wmma_f32_16x16x4_f32(
        /*neg_a=*/false, a, /*neg_b=*/false, ones,
        /*c_mod=*/(short)0, c, /*reuse_a=*/false, /*reuse_b=*/false);
  }
  float s = c[0] + c[1] + c[2] + c[3] + c[4] + c[5] + c[6] + c[7];
  // lane 0 holds rows 0..7 summed (col 0), lane 16 holds rows 8..15
  return __shfl(s, 0, 32) + __shfl(s, 16, 32);
}

__global__ void k_final(const float* __restrict__ pLoc,
                        const float* __restrict__ pCe,
                        const float* __restrict__ pMask,
                        const float* __restrict__ pHard,
                        const int*   __restrict__ nPos,
                        float* __restrict__ out) {
  const int lane = threadIdx.x;        // exactly 32 threads, EXEC all-1s for WMMA
  float locSum  = wmmaSum1024(pLoc,  lane);
  float cePos   = wmmaSum1024(pCe,   lane);
  float maskSum = wmmaSum1024(pMask, lane);
  float h = (lane < Bc) ? pHard[lane] : 0.f;
#pragma unroll
  for (int o = 16; o; o >>= 1) h += __shfl_down(h, o, 32);
  int np = (lane < Bc) ? nPos[lane] : 0;
#pragma unroll
  for (int o = 16; o; o >>= 1) np += __shfl_down(np, o, 32);
  if (lane == 0) {
    float nposf = (float)np;
    float conf = (h + cePos) / nposf;
    float loc  = locSum / fmaxf(nposf * 4.f, 1.f);
    float mask = maskSum / (float)HWc / (float)Bc;
    out[0] = conf + loc + mask;
  }
}

// ---------------- launcher ----------------
extern "C" void kernel_launch(void* const* d_in, const int* in_sizes, int n_in,
                              void* d_out, int out_size, void* d_ws, size_t ws_size,
                              hipStream_t stream) {
  const float*  plocs   = (const float*)d_in[0];
  const float*  pscores = (const float*)d_in[1];
  const float*  pmasks  = (const float*)d_in[2];
  const float*  boxes   = (const float*)d_in[3];
  const float*  priors  = (const float*)d_in[4];
  const int*    labels  = (const int*)d_in[5];
  const int*    masks   = (const int*)d_in[6];
  float* out = (float*)d_out;

  char* ws = (char*)d_ws;
  unsigned long long* best = (unsigned long long*)(ws + OFF_BEST);
  int*   nPos   = (int*)  (ws + OFF_NPOS);
  float* pHard  = (float*)(ws + OFF_PHARD);
  float* pLoc   = (float*)(ws + OFF_PLOC);
  float* pCe    = (float*)(ws + OFF_PCE);
  float* pMask  = (float*)(ws + OFF_PMASK);
  int*   objArr = (int*)  (ws + OFF_OBJ);
  float* ovlArr = (float*)(ws + OFF_OVL);
  float* ceNeg  = (float*)(ws + OFF_CENEG);

  k_init<<<1, 256, 0, stream>>>(best, nPos);
  k_iou<<<dim3(Pc / 256, Bc), 256, 0, stream>>>(
      (const float4*)boxes, (const float4*)priors, best, objArr, ovlArr);
  k_winner<<<1, 256, 0, stream>>>(best, objArr, ovlArr);
  k_prior_loss<<<dim3(Pc / 256, Bc), 256, 0, stream>>>(
      (const float4*)plocs, pscores, (const float4*)boxes,
      (const float4*)priors, labels, objArr, ovlArr,
      ceNeg, nPos, pLoc, pCe);
  k_hardneg<<<Bc, 1024, 0, stream>>>(ceNeg, nPos, pHard);
  k_mask<<<dim3(HWc / 1024, Bc), 512, 0, stream>>>(pmasks, masks, pMask);
  k_final<<<1, 32, 0, stream>>>(pLoc, pCe, pMask, pHard, nPos, out);
}